// CrossDeformAttn_62526133895482
// MI455X (gfx1250) — compile-verified
//
#include <hip/hip_runtime.h>
#include <hip/hip_bf16.h>
#include <math.h>

// ---------------------------------------------------------------------------
// CrossDeformAttn for MI455X (gfx1250, wave32, WMMA)
// B=4, L=4096, C=512, G=4 (GC=128), H=16 (HC=32), K=5
// ---------------------------------------------------------------------------

typedef float v2f_t __attribute__((ext_vector_type(2)));
typedef float v8f_t __attribute__((ext_vector_type(8)));

static constexpr int Bn  = 4;
static constexpr int Ln  = 4096;
static constexpr int Cn  = 512;
static constexpr int GCn = 128;
static constexpr int Hn  = 16;
static constexpr int HCn = 32;
static constexpr int Kc  = 5;
static constexpr int Mtot = Bn * Ln;          // 16384 rows

#define WMMA_F32(a, b, c) \
    __builtin_amdgcn_wmma_f32_16x16x4_f32(false, (a), false, (b), (short)0, (c), false, false)

// ======================= WMMA f32 GEMM =====================================
// Out[M][N] = A[M][K] * W[N][K]^T + bias[N]  (+ rpb[N][Lmod] if rpb != null)
// Block tile 128x64, 8 waves (4 along M, 2 along N), each wave 32x32 via
// four V_WMMA_F32_16X16X4_F32 accumulators (1 LDS v2f load per WMMA).
__global__ __launch_bounds__(256) void gemm_f32_wmma(
    const float* __restrict__ A, const float* __restrict__ W,
    const float* __restrict__ bias, const float* __restrict__ rpb,
    float* __restrict__ Out, int M, int N, int K, int Lmod)
{
    __shared__ float As[128][34];   // +2 pad: conflict-free v2f reads
    __shared__ float Bs[64][34];

    const int tid  = threadIdx.x;
    const int lane = tid & 31;
    const int wave = tid >> 5;      // 0..7
    const int wm   = wave & 3;      // wave M index -> 32 rows each
    const int wn   = wave >> 2;     // wave N index -> 32 cols each
    const int bm   = blockIdx.x * 128;
    const int bn   = blockIdx.y * 64;

    v8f_t acc00 = {}, acc01 = {}, acc10 = {}, acc11 = {};

    const int rl    = lane & 15;
    const int khalf = (lane >> 4) << 1;   // 0 for lanes 0-15, 2 for 16-31

    for (int k0 = 0; k0 < K; k0 += 32) {
        // Stage A tile 128x32 (4 float4/thread) and B tile 64x32 (2/thread)
        #pragma unroll
        for (int i = 0; i < 4; ++i) {
            int idx = tid + i * 256;          // 0..1023
            int r   = idx >> 3;               // row 0..127
            int kcc = (idx & 7) << 2;         // k offset 0..28
            float4 va = *reinterpret_cast<const float4*>(
                &A[(size_t)(bm + r) * K + k0 + kcc]);
            As[r][kcc + 0] = va.x; As[r][kcc + 1] = va.y;
            As[r][kcc + 2] = va.z; As[r][kcc + 3] = va.w;
        }
        #pragma unroll
        for (int i = 0; i < 2; ++i) {
            int idx = tid + i * 256;
            int r   = idx >> 3;
            int kcc = (idx & 7) << 2;
            float4 vb = *reinterpret_cast<const float4*>(
                &W[(size_t)(bn + r) * K + k0 + kcc]);
            Bs[r][kcc + 0] = vb.x; Bs[r][kcc + 1] = vb.y;
            Bs[r][kcc + 2] = vb.z; Bs[r][kcc + 3] = vb.w;
        }
        __syncthreads();

        #pragma unroll
        for (int kk = 0; kk < 32; kk += 4) {
            // A frags: lanes 0-15 hold (K0,K1) of row M, lanes 16-31 (K2,K3)
            v2f_t a0 = *reinterpret_cast<const v2f_t*>(&As[wm * 32 +      rl][kk + khalf]);
            v2f_t a1 = *reinterpret_cast<const v2f_t*>(&As[wm * 32 + 16 + rl][kk + khalf]);
            // B frags: lane -> column N, lane halves split K the same way
            v2f_t b0 = *reinterpret_cast<const v2f_t*>(&Bs[wn * 32 +      rl][kk + khalf]);
            v2f_t b1 = *reinterpret_cast<const v2f_t*>(&Bs[wn * 32 + 16 + rl][kk + khalf]);
            acc00 = WMMA_F32(a0, b0, acc00);
            acc01 = WMMA_F32(a0, b1, acc01);
            acc10 = WMMA_F32(a1, b0, acc10);
            acc11 = WMMA_F32(a1, b1, acc11);
        }
        __syncthreads();
    }

    // Epilogue: C/D layout -> VGPR r: lanes 0-15 M=r, lanes 16-31 M=r+8
    const int mh = (lane >> 4) << 3;
    v8f_t accs[2][2] = {{acc00, acc01}, {acc10, acc11}};
    #pragma unroll
    for (int mi = 0; mi < 2; ++mi) {
        #pragma unroll
        for (int ni = 0; ni < 2; ++ni) {
            #pragma unroll
            for (int r = 0; r < 8; ++r) {
                int m = bm + wm * 32 + mi * 16 + r + mh;
                int n = bn + wn * 32 + ni * 16 + rl;
                float v = accs[mi][ni][r] + bias[n];
                if (rpb) {
                    int l = m % Lmod;   // m = b*L + l
                    v += rpb[(size_t)n * Lmod + l];
                }
                Out[(size_t)m * N + n] = v;
            }
        }
    }
}

// ================= conv1 (k=5, GC->GC, pad=2) as implicit WMMA GEMM ========
// off1[bg][do][l] = boff1[do] + sum_{t,di} Woff1[do][di][t]*qg[bg][di][l+t-2]
// Implicit GEMM: M=do(128), N=l(64 per block), K' = (t,di) = 640.
// A operand = weights wS[do][t][di]; B operand = shifted input window,
// staged transposed qT[pos][di] so B fragments are contiguous v2f.
__global__ __launch_bounds__(256) void conv1_wmma_kernel(
    const float* __restrict__ Qt, const float* __restrict__ Woff1,
    const float* __restrict__ boff1, float* __restrict__ off1)
{
    const int bg = blockIdx.y;            // 0..15
    const int b  = bg >> 2, g = bg & 3;
    const int l0 = blockIdx.x * 64;       // 64 l-blocks

    __shared__ float wS[128][5][10];      // [do][t][di(8, pad 10)]  25.6 KB
    __shared__ float qT[68][10];          // [l0-2 .. l0+65][di]      2.7 KB

    const int tid  = threadIdx.x;
    const int lane = tid & 31;
    const int wave = tid >> 5;
    const int wm   = wave & 3;            // 32 do rows each
    const int wn   = wave >> 2;           // 32 l cols each
    const int rl    = lane & 15;
    const int khalf = (lane >> 4) << 1;

    v8f_t acc00 = {}, acc01 = {}, acc10 = {}, acc11 = {};

    for (int di0 = 0; di0 < GCn; di0 += 8) {
        // stage weights: 128 do x 5 t x 8 di = 5120 (20/thread)
        for (int t = tid; t < 5120; t += 256) {
            int dd = t / 40; int rem = t - dd * 40;
            int tt = rem >> 3; int di = rem & 7;
            wS[dd][tt][di] = Woff1[((size_t)dd * GCn + di0 + di) * Kc + tt];
        }
        // stage input transposed: pos 0..67 -> l = l0 + pos - 2 (zero pad)
        for (int t = tid; t < 68 * 8; t += 256) {
            int pos = t >> 3; int di = t & 7;
            int l = l0 + pos - 2;
            float v = 0.f;
            if (l >= 0 && l < Ln)
                v = Qt[((size_t)(b * Ln + l)) * Cn + g * GCn + di0 + di];
            qT[pos][di] = v;
        }
        __syncthreads();

        #pragma unroll
        for (int tt = 0; tt < 5; ++tt) {
            #pragma unroll
            for (int di4 = 0; di4 < 8; di4 += 4) {
                v2f_t a0 = *reinterpret_cast<const v2f_t*>(
                    &wS[wm * 32 +      rl][tt][di4 + khalf]);
                v2f_t a1 = *reinterpret_cast<const v2f_t*>(
                    &wS[wm * 32 + 16 + rl][tt][di4 + khalf]);
                // B[k][n] = qg[di0+di4+k][l0 + n + tt - 2] = qT[n + tt][di4 + k]
                v2f_t b0 = *reinterpret_cast<const v2f_t*>(
                    &qT[wn * 32 +      rl + tt][di4 + khalf]);
                v2f_t b1 = *reinterpret_cast<const v2f_t*>(
                    &qT[wn * 32 + 16 + rl + tt][di4 + khalf]);
                acc00 = WMMA_F32(a0, b0, acc00);
                acc01 = WMMA_F32(a0, b1, acc01);
                acc10 = WMMA_F32(a1, b0, acc10);
                acc11 = WMMA_F32(a1, b1, acc11);
            }
        }
        __syncthreads();
    }

    const int mh = (lane >> 4) << 3;
    v8f_t accs[2][2] = {{acc00, acc01}, {acc10, acc11}};
    #pragma unroll
    for (int mi = 0; mi < 2; ++mi) {
        #pragma unroll
        for (int ni = 0; ni < 2; ++ni) {
            #pragma unroll
            for (int r = 0; r < 8; ++r) {
                int dd = wm * 32 + mi * 16 + r + mh;
                int l  = l0 + wn * 32 + ni * 16 + rl;
                off1[((size_t)bg * GCn + dd) * Ln + l] = accs[mi][ni][r] + boff1[dd];
            }
        }
    }
}

// ============== conv2 (1x1, pad=2) + tanh + grid index =====================
// off = tanh(boff2 + sum_di Woff2[di]*off1[bg][di][p-2]) * 5
// vgrid = p + off ; iy = vgrid * (L/(L+3)) - 0.5 ; i0=floor(iy); w1=iy-i0
__global__ __launch_bounds__(256) void conv2_idx_kernel(
    const float* __restrict__ off1, const float* __restrict__ Woff2,
    const float* __restrict__ boff2, int* __restrict__ i0buf,
    float* __restrict__ w1buf)
{
    const int bg = blockIdx.y;
    const int p  = blockIdx.x * 256 + threadIdx.x;   // 0..4095

    float s = 0.f;
    const int q = p - 2;
    if (q >= 0) {   // q <= 4093 always for p < 4096
        const float* base = &off1[(size_t)bg * GCn * Ln + q];
        #pragma unroll 4
        for (int di = 0; di < GCn; ++di)
            s += Woff2[di] * base[(size_t)di * Ln];
    }
    s += boff2[0];
    float off   = tanhf(s) * (float)Kc;
    float vgrid = (float)p + off;
    float iy    = vgrid * ((float)Ln / (float)(Ln + 4 - 1)) - 0.5f;
    float f     = floorf(iy);
    i0buf[bg * Ln + p] = (int)f;
    w1buf[bg * Ln + p] = iy - f;
}

// ======================= bilinear gather ===================================
// xs[(b*L+p)*C + g*GC + d] = x[b][i0][g*GC+d]*(1-w1) + x[b][i0+1][...]*w1
// zeros outside [0,L). One wave per p; 32 lanes x float4 = 128 channels.
__global__ __launch_bounds__(256) void sample_kernel(
    const float* __restrict__ x, const int* __restrict__ i0buf,
    const float* __restrict__ w1buf, float* __restrict__ xs)
{
    const int bg   = blockIdx.y;
    const int b    = bg >> 2, g = bg & 3;
    const int p    = blockIdx.x * 8 + (threadIdx.x >> 5);
    const int lane = threadIdx.x & 31;

    const int   i0 = i0buf[bg * Ln + p];
    const float w1 = w1buf[bg * Ln + p];
    const int   c  = g * GCn + lane * 4;

    float4 v0 = make_float4(0.f, 0.f, 0.f, 0.f);
    float4 v1 = v0;
    if (i0 >= 0 && i0 < Ln)
        v0 = *reinterpret_cast<const float4*>(&x[((size_t)(b * Ln + i0)) * Cn + c]);
    if (i0 + 1 >= 0 && i0 + 1 < Ln)
        v1 = *reinterpret_cast<const float4*>(&x[((size_t)(b * Ln + i0 + 1)) * Cn + c]);

    const float w0 = 1.f - w1;
    float4 r;
    r.x = v0.x * w0 + v1.x * w1;
    r.y = v0.y * w0 + v1.y * w1;
    r.z = v0.z * w0 + v1.z * w1;
    r.w = v0.w * w0 + v1.w * w1;
    *reinterpret_cast<float4*>(&xs[((size_t)(b * Ln + p)) * Cn + c]) = r;
}

// ======================= channel attention per head ========================
// logits[i][j] = sum_l q[i][l]k[j][l] * C^-0.5 ; softmax rows ; o = attn @ v
// Qt/Kt/Vt are [b*L+l][C]; head h uses channels h*32..h*32+31.
// Ot written as [b*L+l][h*32+i]  (== reference's final (b,l,h,i) layout)
__global__ __launch_bounds__(256) void attn_kernel(
    const float* __restrict__ Qt, const float* __restrict__ Kt,
    const float* __restrict__ Vt, float* __restrict__ Ot, float scale)
{
    const int bh = blockIdx.x;         // 0..63
    const int b  = bh >> 4, h = bh & 15;
    const int tid = threadIdx.x;

    __shared__ float qs[64][33];
    __shared__ float ks[64][33];
    __shared__ float S[32][33];

    const int i  = tid >> 3;           // 0..31
    const int j0 = (tid & 7) << 2;     // 0,4,..,28
    float accS[4] = {0.f, 0.f, 0.f, 0.f};

    for (int l0 = 0; l0 < Ln; l0 += 64) {
        for (int t = tid; t < 2048; t += 256) {
            int l = t >> 5, c = t & 31;
            size_t idx = ((size_t)(b * Ln + l0 + l)) * Cn + h * HCn + c;
            qs[l][c] = Qt[idx];
            ks[l][c] = Kt[idx];
        }
        __syncthreads();
        #pragma unroll 4
        for (int l = 0; l < 64; ++l) {
            float qv = qs[l][i];
            accS[0] += qv * ks[l][j0 + 0];
            accS[1] += qv * ks[l][j0 + 1];
            accS[2] += qv * ks[l][j0 + 2];
            accS[3] += qv * ks[l][j0 + 3];
        }
        __syncthreads();
    }
    #pragma unroll
    for (int u = 0; u < 4; ++u) S[i][j0 + u] = accS[u] * scale;
    __syncthreads();

    if (tid < 32) {                    // softmax, one lane per row
        float mx = -1e30f;
        for (int j = 0; j < 32; ++j) mx = fmaxf(mx, S[tid][j]);
        float sum = 0.f;
        for (int j = 0; j < 32; ++j) {
            float e = __expf(S[tid][j] - mx);
            S[tid][j] = e; sum += e;
        }
        float inv = 1.f / sum;
        for (int j = 0; j < 32; ++j) S[tid][j] *= inv;
    }
    __syncthreads();

    const int ii    = tid & 31;
    const int lbase = (tid >> 5) << 3;
    for (int l0 = 0; l0 < Ln; l0 += 64) {
        for (int t = tid; t < 2048; t += 256) {
            int l = t >> 5, c = t & 31;
            qs[l][c] = Vt[((size_t)(b * Ln + l0 + l)) * Cn + h * HCn + c];
        }
        __syncthreads();
        #pragma unroll
        for (int u = 0; u < 8; ++u) {
            int l = lbase + u;
            float acc = 0.f;
            #pragma unroll 8
            for (int j = 0; j < 32; ++j) acc += S[ii][j] * qs[l][j];
            Ot[((size_t)(b * Ln + l0 + l)) * Cn + h * HCn + ii] = acc;
        }
        __syncthreads();
    }
}

// ======================= launch ============================================
extern "C" void kernel_launch(void* const* d_in, const int* in_sizes, int n_in,
                              void* d_out, int out_size, void* d_ws, size_t ws_size,
                              hipStream_t stream) {
    (void)in_sizes; (void)n_in; (void)out_size; (void)ws_size;

    const float* x     = (const float*)d_in[0];
    const float* Wq    = (const float*)d_in[1];
    const float* bq    = (const float*)d_in[2];
    const float* Wk    = (const float*)d_in[3];
    const float* bk    = (const float*)d_in[4];
    const float* Wv    = (const float*)d_in[5];
    const float* bv    = (const float*)d_in[6];
    const float* Woff1 = (const float*)d_in[7];
    const float* boff1 = (const float*)d_in[8];
    const float* Woff2 = (const float*)d_in[9];
    const float* boff2 = (const float*)d_in[10];
    const float* rpb   = (const float*)d_in[11];
    const float* Wout  = (const float*)d_in[12];
    const float* bout  = (const float*)d_in[13];
    float* out = (float*)d_out;

    const size_t SZ = (size_t)Mtot * Cn;          // 8M floats per buffer
    float* ws   = (float*)d_ws;
    float* Qt   = ws;                             // [M][C]
    float* OFF1 = ws + 1 * SZ;                    // [16][128][L], reused as Ot
    float* XSt  = ws + 2 * SZ;                    // [M][C]
    float* Kt   = ws + 3 * SZ;
    float* Vt   = ws + 4 * SZ;
    int*   I0   = (int*)(ws + 5 * SZ);            // [16][L]
    float* W1   = ws + 5 * SZ + (size_t)16 * Ln;  // [16][L]
    float* Ot   = OFF1;                           // reuse after conv2 consumed

    dim3 blk(256);
    dim3 gemm_grid(Mtot / 128, Cn / 64);          // 128 x 8

    // 1) Q = X * Wq^T + bq
    gemm_f32_wmma<<<gemm_grid, blk, 0, stream>>>(x, Wq, bq, nullptr, Qt,
                                                 Mtot, Cn, Cn, Ln);
    // 2) offset conv (k=5) as implicit WMMA GEMM
    conv1_wmma_kernel<<<dim3(Ln / 64, 16), blk, 0, stream>>>(Qt, Woff1, boff1,
                                                             OFF1);
    // 3) 1x1 conv + tanh + sampling indices
    conv2_idx_kernel<<<dim3(Ln / 256, 16), blk, 0, stream>>>(OFF1, Woff2, boff2,
                                                             I0, W1);
    // 4) bilinear gather -> XS [M][C]
    sample_kernel<<<dim3(Ln / 8, 16), blk, 0, stream>>>(x, I0, W1, XSt);
    // 5) K = XS * Wk^T + bk
    gemm_f32_wmma<<<gemm_grid, blk, 0, stream>>>(XSt, Wk, bk, nullptr, Kt,
                                                 Mtot, Cn, Cn, Ln);
    // 6) V = XS * Wv^T + bv + rpb
    gemm_f32_wmma<<<gemm_grid, blk, 0, stream>>>(XSt, Wv, bv, rpb, Vt,
                                                 Mtot, Cn, Cn, Ln);
    // 7) per-head channel attention -> Ot
    attn_kernel<<<dim3(Bn * Hn), blk, 0, stream>>>(Qt, Kt, Vt, Ot,
                                                   rsqrtf((float)Cn));
    // 8) out = O * Wout^T + bout
    gemm_f32_wmma<<<gemm_grid, blk, 0, stream>>>(Ot, Wout, bout, nullptr, out,
                                                 Mtot, Cn, Cn, Ln);
}